// AmbientSphericalBrownianMotion_49194555409004
// MI455X (gfx1250) — compile-verified
//
#include <hip/hip_runtime.h>
#include <stdint.h>

// ---------------------------------------------------------------------------
// AmbientSphericalBrownianMotion, MI455X (gfx1250, wave32).
//
// Fully fused: each thread integrates its point(s) for all `steps` in VGPRs;
// HBM traffic is one read + one write of the state (96 MB total, ~4 us at
// 23.3 TB/s) vs ~9.6 GB for an unfused scan. The kernel is compute-bound on
// threefry2x32.
//
// Key trick (even N, the reference case): threefry2x32 of the counter pair
// (j, j+half) yields o0 = bits for flat index j and o1 = bits for index
// j+half. With half = 3N/2 divisible by 3, index 3p+c pairs with index
// 3(p+N/2)+c, i.e. coordinate c of point p pairs with coordinate c of point
// p+N/2. A thread owning the pair (p, p+N/2) therefore consumes BOTH outputs
// of every threefry call: 3 calls per 2 points per step instead of 6 —
// ~35% less total VALU work since threefry dominates.
//
// CDNA5 paths: async global->LDS B128 staging (ASYNCcnt + s_wait_asynccnt),
// LDS broadcast of the per-step folded keys (computed once per workgroup),
// wave32-sized blocks. No WMMA: integer RNG + 3-vector geometry has no
// matmul to map onto the matrix pipes.
// ---------------------------------------------------------------------------

#define TPB  256
#define KMAX 512   // max steps whose folded keys are cached in LDS (4 KB)

typedef int v4i __attribute__((ext_vector_type(4)));
typedef __attribute__((address_space(1))) v4i* gp_v4i;   // global v4i*
typedef __attribute__((address_space(3))) v4i* lp_v4i;   // LDS v4i*

// ---- async global -> LDS copy (16 bytes / lane) ---------------------------
__device__ __forceinline__ void async_copy_b128(const void* g, void* l) {
#if __has_builtin(__builtin_amdgcn_global_load_async_to_lds_b128)
  __builtin_amdgcn_global_load_async_to_lds_b128((gp_v4i)g, (lp_v4i)l, 0, 0);
#else
  uint32_t loff = (uint32_t)(uintptr_t)l;
  uint64_t ga   = (uint64_t)(uintptr_t)g;
  asm volatile("global_load_async_to_lds_b128 %0, %1, off"
               :: "v"(loff), "v"(ga) : "memory");
#endif
}

__device__ __forceinline__ void wait_async0() {
#if __has_builtin(__builtin_amdgcn_s_wait_asynccnt)
  __builtin_amdgcn_s_wait_asynccnt(0);
#else
  asm volatile("s_wait_asynccnt 0" ::: "memory");
#endif
}

// ---- JAX threefry2x32 (20 rounds, JAX rotation schedule) ------------------
__device__ __forceinline__ void tf2x32(uint32_t k0, uint32_t k1,
                                       uint32_t x0, uint32_t x1,
                                       uint32_t& o0, uint32_t& o1) {
  const uint32_t k2 = k0 ^ k1 ^ 0x1BD11BDAu;
  x0 += k0; x1 += k1;
#define TF_ROT(r) { x0 += x1; x1 = (x1 << (r)) | (x1 >> (32 - (r))); x1 ^= x0; }
  TF_ROT(13) TF_ROT(15) TF_ROT(26) TF_ROT(6)
  x0 += k1; x1 += k2 + 1u;
  TF_ROT(17) TF_ROT(29) TF_ROT(16) TF_ROT(24)
  x0 += k2; x1 += k0 + 2u;
  TF_ROT(13) TF_ROT(15) TF_ROT(26) TF_ROT(6)
  x0 += k0; x1 += k1 + 3u;
  TF_ROT(17) TF_ROT(29) TF_ROT(16) TF_ROT(24)
  x0 += k1; x1 += k2 + 4u;
  TF_ROT(13) TF_ROT(15) TF_ROT(26) TF_ROT(6)
  x0 += k2; x1 += k0 + 5u;
#undef TF_ROT
  o0 = x0; o1 = x1;
}

// ---- erfinv (Giles 2012 single-precision), branchless select --------------
__device__ __forceinline__ float erfinv_f(float x) {
  float w = -logf((1.0f - x) * (1.0f + x));
  float wa = w - 2.5f;
  float p1 = 2.81022636e-08f;
  p1 = fmaf(p1, wa,  3.43273939e-07f);
  p1 = fmaf(p1, wa, -3.52338770e-06f);
  p1 = fmaf(p1, wa, -4.39150654e-06f);
  p1 = fmaf(p1, wa,  2.18580870e-04f);
  p1 = fmaf(p1, wa, -1.25372503e-03f);
  p1 = fmaf(p1, wa, -4.17768164e-03f);
  p1 = fmaf(p1, wa,  2.46640727e-01f);
  p1 = fmaf(p1, wa,  1.50140941e+00f);
  float wb = sqrtf(w) - 3.0f;
  float p2 = -2.00214257e-04f;
  p2 = fmaf(p2, wb,  1.00950558e-04f);
  p2 = fmaf(p2, wb,  1.34934322e-03f);
  p2 = fmaf(p2, wb, -3.67342844e-03f);
  p2 = fmaf(p2, wb,  5.73950773e-03f);
  p2 = fmaf(p2, wb, -7.62246130e-03f);
  p2 = fmaf(p2, wb,  9.43887047e-03f);
  p2 = fmaf(p2, wb,  1.00167406e+00f);
  p2 = fmaf(p2, wb,  2.83297682e+00f);
  float p = (w < 5.0f) ? p1 : p2;   // rare tail; select avoids wave32 divergence
  return p * x;
}

// raw 32 bits -> N(0,1): JAX uniform in [-0.99999994, 1) then sqrt(2)*erfinv
__device__ __forceinline__ float bits_to_normal(uint32_t bits) {
  float f = __uint_as_float((bits >> 9) | 0x3F800000u) - 1.0f;
  float u = fmaf(f, 2.0f, -0.99999994f);
  u = fmaxf(u, -0.99999994f);
  return 1.41421356f * erfinv_f(u);
}

// one Stratonovich-Heun step + reprojection, state updated in place
__device__ __forceinline__ void heun_step(float& x0, float& x1, float& x2,
                                          float i0, float i1, float i2) {
  const float dot = x0 * i0 + x1 * i1 + x2 * i2;
  const float nn  = x0 * x0 + x1 * x1 + x2 * x2;
  const float c0  = dot / nn;
  const float g00 = i0 - c0 * x0, g01 = i1 - c0 * x1, g02 = i2 - c0 * x2;
  const float y0 = x0 + g00, y1 = x1 + g01, y2 = x2 + g02;
  const float dot2 = y0 * i0 + y1 * i1 + y2 * i2;
  const float nn2  = y0 * y0 + y1 * y1 + y2 * y2;
  const float c1   = dot2 / nn2;
  const float g10 = i0 - c1 * y0, g11 = i1 - c1 * y1, g12 = i2 - c1 * y2;
  const float z0 = x0 + 0.5f * (g00 + g10);
  const float z1 = x1 + 0.5f * (g01 + g11);
  const float z2 = x2 + 0.5f * (g02 + g12);
  const float inv = 1.0f / sqrtf(z0 * z0 + z1 * z1 + z2 * z2);
  x0 = z0 * inv; x1 = z1 * inv; x2 = z2 * inv;
}

// ---------------------------------------------------------------------------
// Paired kernel (N even): thread t owns points p = t and q = t + M (M = N/2).
// Every threefry output is consumed: o0 -> point p, o1 -> point q.
// ---------------------------------------------------------------------------
__global__ __launch_bounds__(TPB) void
sbm_paired_kernel(const float* __restrict__ xin,
                  const int*   __restrict__ tptr,
                  const int*   __restrict__ sptr,
                  float*       __restrict__ out,
                  int M) {                       // M = N/2
  __shared__ __align__(16) float tileA[TPB * 3];
  __shared__ __align__(16) float tileB[TPB * 3];
  __shared__ uint32_t skeys[2 * KMAX];

  const int tid        = (int)threadIdx.x;
  const int blockStart = (int)blockIdx.x * TPB;
  const int t          = blockStart + tid;      // p = t, q = t + M

  const int   steps = *sptr;
  const float dt    = (float)(*tptr) / (float)steps;
  const float sdt   = sqrtf(dt);
  const uint32_t half = 3u * (uint32_t)M;       // = 3N/2, counter pair stride

  // per-step folded keys: fold_in(key(1), s) = threefry2x32({0,1}, {0,s})
  for (int s = tid; s < steps && s < KMAX; s += TPB) {
    uint32_t o0, o1;
    tf2x32(0u, 1u, 0u, (uint32_t)s, o0, o1);
    skeys[2 * s]     = o0;
    skeys[2 * s + 1] = o1;
  }

  // Async-stage both point tiles (2 x 3072 B) on ASYNCcnt: 192 lanes x 16 B each.
  const bool full = (blockStart + TPB) <= M;
  if (full) {
    if (tid < 192) {
      const char* gA = (const char*)(xin + 3 * (size_t)blockStart) + tid * 16;
      const char* gB = (const char*)(xin + 3 * ((size_t)M + blockStart)) + tid * 16;
      async_copy_b128(gA, (char*)tileA + tid * 16);
      async_copy_b128(gB, (char*)tileB + tid * 16);
    }
    wait_async0();
  }
  __syncthreads();

  float a0, a1, a2, b0, b1, b2;
  if (full) {
    a0 = tileA[3 * tid + 0]; a1 = tileA[3 * tid + 1]; a2 = tileA[3 * tid + 2];
    b0 = tileB[3 * tid + 0]; b1 = tileB[3 * tid + 1]; b2 = tileB[3 * tid + 2];
  } else if (t < M) {
    const size_t p = (size_t)t, q = (size_t)t + (size_t)M;
    a0 = xin[3 * p + 0]; a1 = xin[3 * p + 1]; a2 = xin[3 * p + 2];
    b0 = xin[3 * q + 0]; b1 = xin[3 * q + 1]; b2 = xin[3 * q + 2];
  } else {
    a0 = b0 = 1.0f; a1 = a2 = b1 = b2 = 0.0f;   // inert lane
  }

  const uint32_t j0 = 3u * (uint32_t)t;          // always < half for t < M

  for (int s = 0; s < steps; ++s) {
    uint32_t fk0, fk1;
    if (s < KMAX) { fk0 = skeys[2 * s]; fk1 = skeys[2 * s + 1]; }
    else          { tf2x32(0u, 1u, 0u, (uint32_t)s, fk0, fk1); }

    uint32_t oA0, oB0, oA1, oB1, oA2, oB2;
    tf2x32(fk0, fk1, j0 + 0u, j0 + 0u + half, oA0, oB0);
    tf2x32(fk0, fk1, j0 + 1u, j0 + 1u + half, oA1, oB1);
    tf2x32(fk0, fk1, j0 + 2u, j0 + 2u + half, oA2, oB2);

    heun_step(a0, a1, a2, bits_to_normal(oA0) * sdt,
                          bits_to_normal(oA1) * sdt,
                          bits_to_normal(oA2) * sdt);
    heun_step(b0, b1, b2, bits_to_normal(oB0) * sdt,
                          bits_to_normal(oB1) * sdt,
                          bits_to_normal(oB2) * sdt);
  }

  if (t < M) {
    const size_t p = (size_t)t, q = (size_t)t + (size_t)M;
    out[3 * p + 0] = a0; out[3 * p + 1] = a1; out[3 * p + 2] = a2;
    out[3 * q + 0] = b0; out[3 * q + 1] = b1; out[3 * q + 2] = b2;
  }
}

// ---------------------------------------------------------------------------
// Generic fallback (any N, incl. odd): one point per thread, per-coordinate
// half-pairing with JAX's odd-count padding.
// ---------------------------------------------------------------------------
__device__ __forceinline__ float draw_normal(uint32_t k0, uint32_t k1,
                                             uint32_t j, uint32_t half) {
  const bool lo = j < half;
  const uint32_t a = lo ? j : j - half;
  const uint32_t b = lo ? j + half : j;
  uint32_t o0, o1;
  tf2x32(k0, k1, a, b, o0, o1);
  return bits_to_normal(lo ? o0 : o1);
}

__global__ __launch_bounds__(TPB) void
sbm_generic_kernel(const float* __restrict__ xin,
                   const int*   __restrict__ tptr,
                   const int*   __restrict__ sptr,
                   float*       __restrict__ out,
                   int N) {
  __shared__ uint32_t skeys[2 * KMAX];

  const int tid = (int)threadIdx.x;
  const int gid = (int)blockIdx.x * TPB + tid;

  const int   steps = *sptr;
  const float dt    = (float)(*tptr) / (float)steps;
  const float sdt   = sqrtf(dt);
  const uint32_t total = 3u * (uint32_t)N;
  const uint32_t half  = (total + 1u) >> 1;     // JAX pads odd counts

  for (int s = tid; s < steps && s < KMAX; s += TPB) {
    uint32_t o0, o1;
    tf2x32(0u, 1u, 0u, (uint32_t)s, o0, o1);
    skeys[2 * s]     = o0;
    skeys[2 * s + 1] = o1;
  }
  __syncthreads();

  float x0 = 1.0f, x1 = 0.0f, x2 = 0.0f;
  if (gid < N) {
    x0 = xin[3 * (size_t)gid + 0];
    x1 = xin[3 * (size_t)gid + 1];
    x2 = xin[3 * (size_t)gid + 2];
  }

  const uint32_t j0 = 3u * (uint32_t)gid;
  for (int s = 0; s < steps; ++s) {
    uint32_t fk0, fk1;
    if (s < KMAX) { fk0 = skeys[2 * s]; fk1 = skeys[2 * s + 1]; }
    else          { tf2x32(0u, 1u, 0u, (uint32_t)s, fk0, fk1); }
    const float i0 = draw_normal(fk0, fk1, j0 + 0u, half) * sdt;
    const float i1 = draw_normal(fk0, fk1, j0 + 1u, half) * sdt;
    const float i2 = draw_normal(fk0, fk1, j0 + 2u, half) * sdt;
    heun_step(x0, x1, x2, i0, i1, i2);
  }

  if (gid < N) {
    out[3 * (size_t)gid + 0] = x0;
    out[3 * (size_t)gid + 1] = x1;
    out[3 * (size_t)gid + 2] = x2;
  }
}

extern "C" void kernel_launch(void* const* d_in, const int* in_sizes, int n_in,
                              void* d_out, int out_size, void* d_ws, size_t ws_size,
                              hipStream_t stream) {
  (void)n_in; (void)out_size; (void)d_ws; (void)ws_size;
  const float* x     = (const float*)d_in[0];
  const int*   t     = (const int*)d_in[1];
  const int*   steps = (const int*)d_in[2];
  const int N = in_sizes[0] / 3;

  if ((N & 1) == 0) {                    // reference case: N = 4,194,304
    const int M = N >> 1;
    const int grid = (M + TPB - 1) / TPB;
    sbm_paired_kernel<<<grid, TPB, 0, stream>>>(x, t, steps, (float*)d_out, M);
  } else {
    const int grid = (N + TPB - 1) / TPB;
    sbm_generic_kernel<<<grid, TPB, 0, stream>>>(x, t, steps, (float*)d_out, N);
  }
}